// QMixtureC_15685220565670
// MI455X (gfx1250) — compile-verified
//
#include <hip/hip_runtime.h>

// Problem constants (match reference): M=3, T=128, B=64, E=512
#define QM_M 3
#define QM_T 128
#define QM_B 64
#define QM_E 512

// Native clang vector (not HIP_vector_type) so nontemporal builtins accept it.
typedef float v4f __attribute__((ext_vector_type(4)));

// out[t,b,e] = sum_m reps[m,t,b,e] * w[t,b,m]   (complex out, real weights)
// Memory-bound: ~134 MB moved, ~5.8 us floor at 23.3 TB/s.
// One block = one t, two b rows. 256 threads, 4 e-values per thread.
// Single-use streaming data -> non-temporal hints on all bulk loads/stores.
__global__ __launch_bounds__(256) void qmixture_wsum_kernel(
    const float* __restrict__ rre,   // [M,T,B,E]
    const float* __restrict__ rim,   // [M,T,B,E]
    const float* __restrict__ w,     // [B,T,M]
    float* __restrict__ out)         // [T,B,E] complex64 interleaved (re,im)
{
    __shared__ float sw[8];          // 2 b-rows x 3 weights (+pad)

    const int tid = threadIdx.x;
    const int blk = blockIdx.x;              // 0 .. T*B/2-1
    const int t   = blk >> 5;                // / (B/2)
    const int b0  = (blk & 31) << 1;         // first of the two b rows

    // --- Stage the 6 block-uniform weights into LDS via CDNA5 async DMA ---
    // low 32 bits of a flat shared pointer
    // are exactly the wave-relative LDS byte offset (ISA 10.2 aperture map).
    if (tid < 6) {
        const int m  = tid % 3;
        const int bb = b0 + tid / 3;
        const float* g = w + (size_t)bb * (QM_T * QM_M) + (size_t)t * QM_M + m;
        unsigned           lds_off = (unsigned)(unsigned long long)(&sw[tid]);
        unsigned long long gaddr   = (unsigned long long)g;
        asm volatile("global_load_async_to_lds_b32 %0, %1, off"
                     :: "v"(lds_off), "v"(gaddr) : "memory");
    }
    asm volatile("s_wait_asynccnt 0x0" ::: "memory");
    __syncthreads();

    const int half = tid >> 7;               // which of the two b rows
    const int b    = b0 + half;
    const int e0   = (tid & 127) << 2;       // 4 e-values per thread

    const float w0 = sw[half * 3 + 0];
    const float w1 = sw[half * 3 + 1];
    const float w2 = sw[half * 3 + 2];

    const size_t base = (size_t)t * (QM_B * QM_E) + (size_t)b * QM_E + e0;
    const size_t mstr = (size_t)QM_T * QM_B * QM_E;

    // 6 x global_load_b128 th:TH_LOAD_NT (96 B/thread) — pure streaming
    const v4f r0 = __builtin_nontemporal_load((const v4f*)(rre + base));
    const v4f r1 = __builtin_nontemporal_load((const v4f*)(rre + base + mstr));
    const v4f r2 = __builtin_nontemporal_load((const v4f*)(rre + base + 2 * mstr));
    const v4f i0 = __builtin_nontemporal_load((const v4f*)(rim + base));
    const v4f i1 = __builtin_nontemporal_load((const v4f*)(rim + base + mstr));
    const v4f i2 = __builtin_nontemporal_load((const v4f*)(rim + base + 2 * mstr));

    v4f re, im;
    re.x = fmaf(w2, r2.x, fmaf(w1, r1.x, w0 * r0.x));
    re.y = fmaf(w2, r2.y, fmaf(w1, r1.y, w0 * r0.y));
    re.z = fmaf(w2, r2.z, fmaf(w1, r1.z, w0 * r0.z));
    re.w = fmaf(w2, r2.w, fmaf(w1, r1.w, w0 * r0.w));
    im.x = fmaf(w2, i2.x, fmaf(w1, i1.x, w0 * i0.x));
    im.y = fmaf(w2, i2.y, fmaf(w1, i1.y, w0 * i0.y));
    im.z = fmaf(w2, i2.z, fmaf(w1, i1.z, w0 * i0.z));
    im.w = fmaf(w2, i2.w, fmaf(w1, i1.w, w0 * i0.w));

    // complex64 is interleaved (re, im): 8 consecutive floats -> 2 x b128 stores
    v4f o0, o1;
    o0.x = re.x; o0.y = im.x; o0.z = re.y; o0.w = im.y;
    o1.x = re.z; o1.y = im.z; o1.z = re.w; o1.w = im.w;

    v4f* op = (v4f*)(out + 2 * ((size_t)(t * QM_B + b) * QM_E + e0));
    __builtin_nontemporal_store(o0, op);
    __builtin_nontemporal_store(o1, op + 1);
}

extern "C" void kernel_launch(void* const* d_in, const int* in_sizes, int n_in,
                              void* d_out, int out_size, void* d_ws, size_t ws_size,
                              hipStream_t stream) {
    (void)in_sizes; (void)n_in; (void)d_ws; (void)ws_size; (void)out_size;
    const float* rre = (const float*)d_in[0];   // reps_real [M,T,B,E] f32
    const float* rim = (const float*)d_in[1];   // reps_imag [M,T,B,E] f32
    const float* wts = (const float*)d_in[2];   // weights   [B,T,M]   f32
    float*       out = (float*)d_out;           // [T,B,E] complex64 interleaved

    const int blocks = (QM_T * QM_B) / 2;       // 4096
    qmixture_wsum_kernel<<<blocks, 256, 0, stream>>>(rre, rim, wts, out);
}